// HartleyConv2d_59167469470018
// MI455X (gfx1250) — compile-verified
//
#include <hip/hip_runtime.h>
#include <hip/hip_bf16.h>
#include <math.h>
#include <stdint.h>

// ---------------------------------------------------------------------------
// HartleyConv2d for MI455X (gfx1250, wave32).
//   x: [8,64,128,128] f32, weight: [64,64,3,3] f32 -> out: [8,64,128,128] f32
//
// out[b,o,u,v] = 0.5/N^2 * sum_c [ X[b,c,u,v]*(W+Wf)[o,c,u,v]
//                                 + X[b,c,mu,mv]*(W-Wf)[o,c,u,v] ]
// X = H * x_pad * H (separable DHT, H[k,n]=cas(2*pi*k*n/132), symmetric),
// W built from the 3x3 support directly, (mu,mv) = (131-u,131-v).
//
// Stage 1/2: fp32 WMMA GEMMs, shared H tile staged in LDS via
//            global_load_async_to_lds (ASYNCcnt path).
// Stage 3:   per-frequency GEMM  [Xp|Xm](8x128) x [G1^T;G2^T](128x64)
//            via fp32 WMMA from LDS.
// ---------------------------------------------------------------------------

typedef float v2f __attribute__((ext_vector_type(2)));
typedef float v8f __attribute__((ext_vector_type(8)));

#define BATCH   8
#define CH      64
#define HS      128          // spatial size
#define NF      132          // fft_h = fft_w = 128 + 2 + 3 - 1
#define NP      144          // NF padded to multiple of 16 for WMMA tiles
#define SLICES  (BATCH * CH) // 512

#define HP_ELEMS (NP * NP)
#define Y_ELEMS  (SLICES * HS * NP)
#define X_ELEMS  (SLICES * NP * NP)

// --- CDNA5 async global->LDS copies (ASYNCcnt), portable via inline asm ----
__device__ __forceinline__ void async_copy_b32(void* lds_dst, const void* gsrc) {
  unsigned lo = (unsigned)(uintptr_t)lds_dst;   // low 32b of generic LDS ptr
  asm volatile("global_load_async_to_lds_b32 %0, %1, off"
               :: "v"(lo), "v"(gsrc) : "memory");
}
__device__ __forceinline__ void async_copy_b128(void* lds_dst, const void* gsrc) {
  unsigned lo = (unsigned)(uintptr_t)lds_dst;
  asm volatile("global_load_async_to_lds_b128 %0, %1, off"
               :: "v"(lo), "v"(gsrc) : "memory");
}
__device__ __forceinline__ void wait_async0() {
  asm volatile("s_wait_asynccnt 0x0" ::: "memory");
}

// ---------------------------------------------------------------------------
// Kernel 0: Hp[k][n] = cas(2*pi*k*n/132), zero-padded to 144x144.
// ---------------------------------------------------------------------------
__global__ __launch_bounds__(256) void k_hartley_table(float* __restrict__ Hp) {
  int idx = blockIdx.x * 256 + threadIdx.x;
  if (idx >= HP_ELEMS) return;
  int k = idx / NP, n = idx % NP;
  float val = 0.0f;
  if (k < NF && n < NF) {
    int m = (k * n) % NF;
    float th = (float)m * (6.28318530717958647692f / (float)NF);
    val = cosf(th) + sinf(th);
  }
  Hp[idx] = val;
}

// ---------------------------------------------------------------------------
// Kernel 1: DHT along w.  Y[s][i][v] = sum_{j<128} x[s][i][j] * Hp[j+1][v]
// grid=(slice, tn). All 8 waves (=tm) share the H column-tile -> LDS (async).
// ---------------------------------------------------------------------------
__global__ __launch_bounds__(256) void k_dht_w(const float* __restrict__ x,
                                               const float* __restrict__ Hp,
                                               float* __restrict__ Y) {
  __shared__ float sB[HS * 16];           // sB[j][vv] = Hp[j+1][c0+vv], 8 KB
  const int s  = blockIdx.x;
  const int tn = blockIdx.y;              // 0..8
  const int c0 = tn * 16;

  // async-stage the B tile: 512 x b128 (16B each), 16B-aligned rows (NP%4==0)
  for (int g = threadIdx.x; g < 512; g += 256) {
    const int j = g >> 2, q = g & 3;
    async_copy_b128(&sB[g * 4], Hp + (j + 1) * NP + c0 + q * 4);
  }
  wait_async0();
  __syncthreads();

  const int wave = threadIdx.x >> 5;      // = tm, 0..7
  const int lane = threadIdx.x & 31;
  const int half = lane >> 4;             // A/B frag halves: K{0,1} vs K{2,3}
  const int lr   = lane & 15;

  const float* xs = x + (size_t)s * (HS * HS);
  const int row = wave * 16 + lr;         // M
  const int col = c0 + lr;                // N

  v8f acc = {};
  for (int k = 0; k < HS; k += 4) {
    const int ka = k + half * 2;
    v2f a, b;
    a.x = xs[row * HS + ka];
    a.y = xs[row * HS + ka + 1];
    b.x = sB[ka * 16 + lr];
    b.y = sB[(ka + 1) * 16 + lr];
    acc = __builtin_amdgcn_wmma_f32_16x16x4_f32(
        false, a, false, b, (short)0, acc, false, false);
  }

  float* ys = Y + (size_t)s * (HS * NP);
  #pragma unroll
  for (int r = 0; r < 8; ++r) {
    const int m = wave * 16 + r + half * 8;
    ys[m * NP + col] = acc[r];
  }
}

// ---------------------------------------------------------------------------
// Kernel 2: DHT along h.  X[s][u][v] = sum_{i<128} Hp[u][i+1] * Y[s][i][v]
// grid=(slice, tm). All 9 waves (=tn) share the H row-tile -> LDS (async).
// Stored column-major so A-frag lane reads hit distinct banks.
// ---------------------------------------------------------------------------
__global__ __launch_bounds__(288) void k_dht_h(const float* __restrict__ Y,
                                               const float* __restrict__ Hp,
                                               float* __restrict__ X) {
  __shared__ float sA[HS * 16];           // sA[i][r] = Hp[tm*16+r][i+1], 8 KB
  const int s  = blockIdx.x;
  const int tm = blockIdx.y;              // 0..8 (u tile; rows >=132 are zero)

  for (int e = threadIdx.x; e < HS * 16; e += 288) {
    const int r = e & 15, i = e >> 4;
    async_copy_b32(&sA[e], Hp + (tm * 16 + r) * NP + i + 1);
  }
  wait_async0();
  __syncthreads();

  const int wave = threadIdx.x >> 5;      // = tn, 0..8
  const int lane = threadIdx.x & 31;
  const int half = lane >> 4;
  const int lr   = lane & 15;

  const float* ys = Y + (size_t)s * (HS * NP);
  const int col = wave * 16 + lr;         // v

  v8f acc = {};
  for (int k = 0; k < HS; k += 4) {
    const int ka = k + half * 2;
    v2f a, b;
    a.x = sA[ka * 16 + lr];
    a.y = sA[(ka + 1) * 16 + lr];
    b.x = ys[ka * NP + col];
    b.y = ys[(ka + 1) * NP + col];
    acc = __builtin_amdgcn_wmma_f32_16x16x4_f32(
        false, a, false, b, (short)0, acc, false, false);
  }

  float* xsl = X + (size_t)s * (NP * NP);
  #pragma unroll
  for (int r = 0; r < 8; ++r) {
    const int m = tm * 16 + r + half * 8; // u
    xsl[m * NP + col] = acc[r];
  }
}

// ---------------------------------------------------------------------------
// Kernel 3: per-frequency combine as a WMMA GEMM. One 128-thread block per
// (u,v):  A(16x128) = [Xp|Xm] (rows 8..15 zero),  B(128x64) = [G1^T ; G2^T],
//   G1[o][c] = sum_ij w[o,c,i,j]*(hu_i*hv_j + hmu_i*hmv_j)
//   G2[o][c] = sum_ij w[o,c,i,j]*(hu_i*hv_j - hmu_i*hmv_j)
// out[b,o] = scale * (A x B)[b][o].  4 waves, each one 16-wide N tile.
// ---------------------------------------------------------------------------
__global__ __launch_bounds__(128) void k_combine(const float* __restrict__ X,
                                                 const float* __restrict__ Hp,
                                                 const float* __restrict__ w,
                                                 float* __restrict__ out) {
  __shared__ float sGT[128 * CH];         // B matrix: sGT[k][o], 32 KB
  __shared__ float sA[BATCH * 128];       // A rows 0..7: [b][k], 4 KB

  const int pos = blockIdx.x;             // 0..16383
  const int u = pos >> 7, v = pos & 127;
  const int mu = (NF - 1) - u, mv = (NF - 1) - v;
  const int tid = threadIdx.x;            // 0..127

  // cas factors for the 3x3 support (uniform across block)
  float a9[9], b9[9];
  #pragma unroll
  for (int i = 0; i < 3; ++i) {
    const float hu  = Hp[u  * NP + i];
    const float hmu = Hp[mu * NP + i];
    #pragma unroll
    for (int j = 0; j < 3; ++j) {
      const float hv  = Hp[v  * NP + j];
      const float hmv = Hp[mv * NP + j];
      const float t1 = hu * hv, t2 = hmu * hmv;
      a9[i * 3 + j] = t1 + t2;
      b9[i * 3 + j] = t1 - t2;
    }
  }

  // stage A: k<64 -> Xp sample, k>=64 -> mirrored sample
  for (int e = tid; e < BATCH * 128; e += 128) {
    const int b = e >> 7, k = e & 127, c = k & 63;
    const float* xsl = X + (size_t)(b * CH + c) * (NP * NP);
    sA[e] = (k < 64) ? xsl[u * NP + v] : xsl[mu * NP + mv];
  }

  // build B = [G1^T ; G2^T]
  for (int pr = tid; pr < CH * CH; pr += 128) {  // pr = o*64 + c
    const int o = pr >> 6, c = pr & 63;
    const float* wp = w + (size_t)pr * 9;
    float g1 = 0.0f, g2 = 0.0f;
    #pragma unroll
    for (int q = 0; q < 9; ++q) {
      g1 = fmaf(wp[q], a9[q], g1);
      g2 = fmaf(wp[q], b9[q], g2);
    }
    sGT[c * CH + o]        = g1;
    sGT[(64 + c) * CH + o] = g2;
  }
  __syncthreads();

  const int wave = tid >> 5;              // 0..3 -> N tile
  const int lane = tid & 31;
  const int half = lane >> 4;
  const int lr   = lane & 15;
  const int n0   = wave * 16;

  v8f acc = {};
  for (int k = 0; k < 128; k += 4) {
    const int ka = k + half * 2;
    v2f a, b;
    a.x = (lr < 8) ? sA[lr * 128 + ka]     : 0.0f;   // rows 8..15 zero-pad
    a.y = (lr < 8) ? sA[lr * 128 + ka + 1] : 0.0f;
    b.x = sGT[ka * CH + n0 + lr];
    b.y = sGT[(ka + 1) * CH + n0 + lr];
    acc = __builtin_amdgcn_wmma_f32_16x16x4_f32(
        false, a, false, b, (short)0, acc, false, false);
  }

  const float scale = 0.5f / ((float)NF * (float)NF);
  if (half == 0) {                        // lanes 0..15 hold rows M=0..7 (=b)
    const int o = n0 + lr;
    #pragma unroll
    for (int r = 0; r < 8; ++r) {         // b = r
      out[(((size_t)r * CH + o) * HS + u) * HS + v] = acc[r] * scale;
    }
  }
}

// ---------------------------------------------------------------------------
// launch
// ---------------------------------------------------------------------------
extern "C" void kernel_launch(void* const* d_in, const int* in_sizes, int n_in,
                              void* d_out, int out_size, void* d_ws, size_t ws_size,
                              hipStream_t stream) {
  const float* x = (const float*)d_in[0];   // [8,64,128,128]
  const float* w = (const float*)d_in[1];   // [64,64,3,3]
  float* out = (float*)d_out;               // [8,64,128,128]

  float* Hp = (float*)d_ws;                 // ~81 KB
  float* Y  = Hp + HP_ELEMS;                // ~37.7 MB
  float* X  = Y + Y_ELEMS;                  // ~42.5 MB
  (void)in_sizes; (void)n_in; (void)out_size; (void)ws_size;

  k_hartley_table<<<(HP_ELEMS + 255) / 256, 256, 0, stream>>>(Hp);
  k_dht_w<<<dim3(SLICES, 9), 256, 0, stream>>>(x, Hp, Y);
  k_dht_h<<<dim3(SLICES, 9), 288, 0, stream>>>(Y, Hp, X);
  k_combine<<<HS * HS, 128, 0, stream>>>(X, Hp, w, out);
}